// Conv2d_53334903882372
// MI455X (gfx1250) — compile-verified
//
#include <hip/hip_runtime.h>

typedef __attribute__((ext_vector_type(2))) float v2f;
typedef __attribute__((ext_vector_type(8))) float v8f;

#define CIN     8
#define COUT    8
#define HW      1024
#define OUT_HW  1022
#define XCHUNK  8            // output rows per block (8 independent WMMA accumulators)
#define ROWS    (XCHUNK + 2) // input rows staged (halo of 1 on each side)
#define COLS    130          // 128 output cols + 2 halo
#define CSTR    132          // padded LDS column stride (bank-conflict padding)
#define KTOT    72           // CIN * 9
#define KSTEPS  18           // KTOT / 4

// Compile-time LDS offset for reduction index K = cin*9 + fi*3 + fj.
__host__ __device__ constexpr int off_of_K(int K) {
    int ci = K / 9;
    int r  = K - ci * 9;
    int fi = r / 3;
    int fj = r - fi * 3;
    return (ci * ROWS + fi) * CSTR + fj;
}

// Implicit-GEMM conv via V_WMMA_F32_16X16X4_F32.
//   M = output channel (16 rows, 8 valid), N = 16 consecutive W pixels,
//   K = cin*9 + fi*3 + fj, accumulated 4 at a time.
// kk outer / xi inner: 8 independent accumulator chains -> no back-to-back
// WMMA RAW hazard; xi offsets fold into ds_load immediate offsets (xi*528).
__global__ __launch_bounds__(256)
void conv3x3_wmma_f32(const float* __restrict__ inp,
                      const float* __restrict__ filt,
                      const float* __restrict__ bias,
                      float* __restrict__ out)
{
    __shared__ float lds[CIN * ROWS * CSTR]; // 8*10*132*4 = 42,240 B

    const int ytile = blockIdx.x;   // 0..7   -> y0 = ytile*128
    const int xtile = blockIdx.y;   // 0..127 -> x0 = xtile*8
    const int n     = blockIdx.z;   // batch

    const int y0 = ytile * 128;
    const int x0 = xtile * XCHUNK;

    const int tid  = threadIdx.x;
    const int lane = tid & 31;
    const int wid  = tid >> 5;
    const int m    = lane & 15;  // row/col-within-16 for both A and B fragments
    const int half = lane >> 4;  // selects K pair {0,1} vs {2,3}

    // ---- stage input slab into LDS ----
    // Wave `wid` owns input channel `wid`; lanes stride columns by 32
    // -> coalesced 128B transactions, no integer division.
    {
        const float* src = inp + ((size_t)(n * CIN + wid)) * HW * HW;
        #pragma unroll
        for (int row = 0; row < ROWS; ++row) {
            const int grow = (x0 - 1 + row + HW) & (HW - 1); // wrap of the -1 row
            const float* srow = src + (size_t)grow * HW;
            float* drow = &lds[(wid * ROWS + row) * CSTR];
            #pragma unroll
            for (int cb = 0; cb < 5; ++cb) {                 // 5*32 = 160 >= 130
                int col = cb * 32 + lane;
                if (col < COLS) {
                    int gcol = y0 + col;
                    if (gcol > HW - 1) gcol = HW - 1;        // clamp right-edge halo
                    drow[col] = srow[gcol];
                }
            }
        }
    }

    // ---- A fragments (filter) and per-lane LDS offsets for B: division-free ----
    // filt index (m*CIN+ci)*9 + fi*3 + fj == m*72 + K, with K = 4*kk + 2*half + j.
    v2f a[KSTEPS];
    int boff[KSTEPS][2];
    const int base  = wid * 16 + m;        // column within the block's 130-wide slab
    const int fbase = m * KTOT + 2 * half; // filter flat base for this lane
    #pragma unroll
    for (int kk = 0; kk < KSTEPS; ++kk) {
        #pragma unroll
        for (int j = 0; j < 2; ++j) {
            float w = (m < COUT) ? filt[fbase + 4 * kk + j] : 0.0f;
            if (j == 0) a[kk].x = w; else a[kk].y = w;
            // select between the two compile-time offsets (half=0 vs half=1)
            boff[kk][j] = (half ? off_of_K(4 * kk + 2 + j)
                                : off_of_K(4 * kk + j)) + base;
        }
    }

    // ---- 8 accumulators seeded with bias ----
    v8f c[XCHUNK];
    {
        float bv[COUT];
        #pragma unroll
        for (int v = 0; v < COUT; ++v) bv[v] = bias[v]; // uniform scalar loads
        #pragma unroll
        for (int xi = 0; xi < XCHUNK; ++xi)
            #pragma unroll
            for (int v = 0; v < 8; ++v) c[xi][v] = bv[v];
    }

    __syncthreads();

    // ---- main MMA: kk outer, xi inner (independent chains) ----
    #pragma unroll
    for (int kk = 0; kk < KSTEPS; ++kk) {
        const float* p0 = &lds[boff[kk][0]];
        const float* p1 = &lds[boff[kk][1]];
        #pragma unroll
        for (int xi = 0; xi < XCHUNK; ++xi) {
            v2f b;
            b.x = p0[xi * CSTR];   // folds to ds_load immediate offset (xi*528)
            b.y = p1[xi * CSTR];
            // 8 args: (neg_a, A, neg_b, B, c_mod, C, reuse_a, reuse_b)
            c[xi] = __builtin_amdgcn_wmma_f32_16x16x4_f32(
                        false, a[kk], false, b, (short)0, c[xi], false, false);
        }
    }

    // ---- epilogue: D layout VGPR v, lanes 0-15 -> (M=v, N=lane) ----
    // channel v, 16 contiguous pixels -> coalesced 64B stores.
    const int y = y0 + base;
    #pragma unroll
    for (int xi = 0; xi < XCHUNK; ++xi) {
        const int x = x0 + xi;
        if (half == 0 && x < OUT_HW && y < OUT_HW) {
            #pragma unroll
            for (int v = 0; v < COUT; ++v)
                out[((n * COUT + v) * OUT_HW + x) * OUT_HW + y] = c[xi][v];
        }
    }
}

extern "C" void kernel_launch(void* const* d_in, const int* in_sizes, int n_in,
                              void* d_out, int out_size, void* d_ws, size_t ws_size,
                              hipStream_t stream) {
    (void)in_sizes; (void)n_in; (void)d_ws; (void)ws_size; (void)out_size;
    const float* inp  = (const float*)d_in[0];  // [8,8,1024,1024]
    const float* filt = (const float*)d_in[1];  // [8,8,3,3]
    const float* bias = (const float*)d_in[2];  // [8]
    float* out = (float*)d_out;                 // [8,8,1022,1022]

    dim3 grid(8 /*y tiles of 128*/, 128 /*x tiles of 8*/, 8 /*batch*/);
    conv3x3_wmma_f32<<<grid, 256, 0, stream>>>(inp, filt, bias, out);
}